// NMS_12945031430230
// MI455X (gfx1250) — compile-verified
//
#include <hip/hip_runtime.h>
#include <math.h>

// Rotated NMS, 2048 boxes, 20 classes.
// Pipeline:
//   k1 nms_sort_prep : bitonic argsort by (-score, idx) in LDS; emit sorted
//                      centers / norms / radii / areas / labels / corners.
//   k2 nms_pairs     : one wave per 16x16 tile of the upper-triangular pair
//                      matrix. V_WMMA_F32_16X16X4_F32 computes the 16x16 Gram
//                      tile ci.cj (K=2 of 4 used); epilogue prunes by label +
//                      center distance, exact rotated-IoU via Sutherland-
//                      Hodgman clipping for survivors; wave32 ballot packs
//                      suppression bits into exclusive u16 words.
//   k3 nms_scan      : sequential greedy keep-scan over the bit matrix
//                      (128 threads own 16 columns each), scatter scores*keep.

typedef __attribute__((ext_vector_type(2))) float v2f;
typedef __attribute__((ext_vector_type(8))) float v8f;

#define NB 2048
#define NTILE 128  // NB/16 tile words per matrix row

// ---------------------------------------------------------------- workspace
// byte offsets into d_ws
#define OFF_ORDER 0u
#define OFF_CX    (OFF_ORDER + NB * 4u)
#define OFF_CY    (OFF_CX + NB * 4u)
#define OFF_NORM  (OFF_CY + NB * 4u)
#define OFF_RAD   (OFF_NORM + NB * 4u)
#define OFF_AREA  (OFF_RAD + NB * 4u)
#define OFF_LAB   (OFF_AREA + NB * 4u)
#define OFF_CORN  (OFF_LAB + NB * 4u)            // NB*8 floats
#define OFF_MAT   (OFF_CORN + NB * 8u * 4u)      // NB*NTILE u16 = 512 KB

// ---------------------------------------------------------------- kernel 1
__global__ __launch_bounds__(1024)
void nms_sort_prep(const float* __restrict__ boxes, const float* __restrict__ scores,
                   const int* __restrict__ labels,
                   unsigned* __restrict__ order, float* __restrict__ cxs,
                   float* __restrict__ cys, float* __restrict__ nrm,
                   float* __restrict__ rad, float* __restrict__ area,
                   int* __restrict__ lab, float* __restrict__ corn) {
  __shared__ unsigned long long key[NB];
  const int tid = threadIdx.x;
  // key = (~scorebits, idx): ascending u64 sort == (score desc, idx asc),
  // matching stable argsort(-scores). scores are non-negative floats.
  for (int i = tid; i < NB; i += 1024) {
    unsigned sb = __float_as_uint(scores[i]);
    key[i] = (((unsigned long long)(~sb)) << 32) | (unsigned)i;
  }
  __syncthreads();
  for (int k = 2; k <= NB; k <<= 1) {
    for (int j = k >> 1; j > 0; j >>= 1) {
      for (int t = tid; t < NB; t += 1024) {
        int ixj = t ^ j;
        if (ixj > t) {
          unsigned long long a = key[t], b = key[ixj];
          bool up = ((t & k) == 0);
          if ((a > b) == up) { key[t] = b; key[ixj] = a; }
        }
      }
      __syncthreads();
    }
  }
  for (int i = tid; i < NB; i += 1024) {
    unsigned o = (unsigned)(key[i] & 0xFFFFFFFFull);
    order[i] = o;
    float cx = boxes[o * 5 + 0], cy = boxes[o * 5 + 1];
    float w = boxes[o * 5 + 2], h = boxes[o * 5 + 3], ang = boxes[o * 5 + 4];
    float c = cosf(ang), s = sinf(ang);
    cxs[i] = cx; cys[i] = cy;
    nrm[i] = cx * cx + cy * cy;
    rad[i] = 0.5f * sqrtf(w * w + h * h);
    area[i] = w * h;
    lab[i] = labels[o];
    float hw = 0.5f * w, hh = 0.5f * h;
    const float dxs[4] = {-hw, hw, hw, -hw};
    const float dys[4] = {-hh, -hh, hh, hh};
    for (int q = 0; q < 4; ++q) {  // CCW corners, as in reference
      corn[i * 8 + 2 * q]     = cx + c * dxs[q] - s * dys[q];
      corn[i * 8 + 2 * q + 1] = cy + s * dxs[q] + c * dys[q];
    }
  }
}

// ------------------------------------------------- exact intersection area
// Sutherland-Hodgman clip of CCW rect A by CCW rect B, in box-A-local
// coordinates (ox,oy) for f32 precision (mirrors reference's pts-center).
__device__ __forceinline__ float polyInterArea(const float* __restrict__ corn,
                                               int gi, int gj, float ox, float oy) {
  float px[12], py[12];
  float bx[4], by[4];
  const float* A = corn + gi * 8;
  const float* B = corn + gj * 8;
  for (int k = 0; k < 4; ++k) {
    px[k] = A[2 * k] - ox;     py[k] = A[2 * k + 1] - oy;
    bx[k] = B[2 * k] - ox;     by[k] = B[2 * k + 1] - oy;
  }
  int n = 4;
  for (int e = 0; e < 4 && n > 0; ++e) {
    float qx = bx[e], qy = by[e];
    int e1 = (e + 1) & 3;
    float rx = bx[e1] - qx, ry = by[e1] - qy;
    float tx[12], ty[12];
    int m = 0;
    for (int k = 0; k < n; ++k) {
      int k1 = (k + 1 == n) ? 0 : k + 1;
      float ax = px[k], ay = py[k], b2x = px[k1], b2y = py[k1];
      float da = rx * (ay - qy) - ry * (ax - qx);   // >=0 == inside (CCW)
      float db = rx * (b2y - qy) - ry * (b2x - qx);
      if (da >= 0.f) { tx[m] = ax; ty[m] = ay; ++m; }
      if ((da >= 0.f) != (db >= 0.f)) {
        float t = da / (da - db);
        tx[m] = ax + t * (b2x - ax); ty[m] = ay + t * (b2y - ay); ++m;
      }
    }
    n = m;
    for (int k = 0; k < n; ++k) { px[k] = tx[k]; py[k] = ty[k]; }
  }
  if (n < 3) return 0.f;
  float s = 0.f;
  for (int k = 0; k < n; ++k) {
    int k1 = (k + 1 == n) ? 0 : k + 1;
    s += px[k] * py[k1] - px[k1] * py[k];
  }
  return 0.5f * fabsf(s);
}

// ---------------------------------------------------------------- kernel 2
__global__ __launch_bounds__(32)
void nms_pairs(const float* __restrict__ cxs, const float* __restrict__ cys,
               const float* __restrict__ nrm, const float* __restrict__ rad,
               const float* __restrict__ area, const int* __restrict__ lab,
               const float* __restrict__ corn, const float* __restrict__ thr_p,
               unsigned short* __restrict__ mat) {
  const int tj = blockIdx.x;  // column tile (j = suppressed candidates)
  const int ti = blockIdx.y;  // row tile    (i = suppressors)
  if (tj < ti) return;        // upper triangle only (uniform exit, EXEC stays full)

  const int lane = threadIdx.x;
  const int half = lane >> 4;   // 0: rows 0..7, 1: rows 8..15 of D tile
  const int n    = lane & 15;   // column within tile

  float dr[8];
#if __has_builtin(__builtin_amdgcn_wmma_f32_16x16x4_f32)
  // A (16x4, v2f/lane): lanes 0-15 hold row M=lane, VGPR0=K0(x), VGPR1=K1(y);
  // lanes 16-31 hold K2,K3 -> zero (K padded 2->4).
  // B (4x16, v2f/lane): mirrored layout, column N=lane for lanes 0-15.
  v2f a = {0.f, 0.f}, b = {0.f, 0.f};
  if (half == 0) {
    a[0] = cxs[ti * 16 + n]; a[1] = cys[ti * 16 + n];
    b[0] = cxs[tj * 16 + n]; b[1] = cys[tj * 16 + n];
  }
  v8f cz = {0.f, 0.f, 0.f, 0.f, 0.f, 0.f, 0.f, 0.f};
  v8f d = __builtin_amdgcn_wmma_f32_16x16x4_f32(
      /*neg_a=*/false, a, /*neg_b=*/false, b,
      /*c_mod=*/(short)0, cz, /*reuse_a=*/false, /*reuse_b=*/false);
  for (int r = 0; r < 8; ++r) dr[r] = d[r];   // element (m = r+8*half, n)
#else
  for (int r = 0; r < 8; ++r) {
    int gi = ti * 16 + r + half * 8;
    dr[r] = cxs[gi] * cxs[tj * 16 + n] + cys[gi] * cys[tj * 16 + n];
  }
#endif

  const float thr = thr_p[0];
  const int gj = tj * 16 + n;
  const int labj = lab[gj];
  const float nj = nrm[gj], rj = rad[gj], aj = area[gj];

  for (int r = 0; r < 8; ++r) {
    const int gi = ti * 16 + r + half * 8;
    bool bit = false;
    if (gj > gi && lab[gi] == labj) {
      // |ci-cj|^2 from Gram tile; raw (un-offset) coords so cancellation error
      // is a few ulp of ~4e6 -> slack 8 covers it plus the reference's 1e-5 eps.
      float d2 = nrm[gi] + nj - 2.f * dr[r];
      float rs = rad[gi] + rj;
      if (d2 <= rs * rs + 8.f) {
        float inter = polyInterArea(corn, gi, gj, cxs[gi], cys[gi]);
        float iou = inter / (area[gi] + aj - inter + 1e-9f);
        bit = iou > thr;
      }
    }
    // wave32 ballot: bits[15:0] = row (r) columns 0..15, bits[31:16] = row (r+8)
    unsigned m32 = (unsigned)__ballot(bit ? 1 : 0);
    if (lane == 0)  mat[(ti * 16 + r) * NTILE + tj]     = (unsigned short)(m32 & 0xFFFFu);
    if (lane == 16) mat[(ti * 16 + r + 8) * NTILE + tj] = (unsigned short)(m32 >> 16);
  }
}

// ---------------------------------------------------------------- kernel 3
__global__ __launch_bounds__(NTILE)
void nms_scan(const unsigned short* __restrict__ mat, const unsigned* __restrict__ order,
              const float* __restrict__ scores, float* __restrict__ out) {
  __shared__ unsigned short keepw[NTILE];
  const int t = threadIdx.x;
  keepw[t] = 0xFFFFu;
  __syncthreads();
  for (int i = 0; i < NB; ++i) {
    // greedy: if box i still kept, clear everything its row suppresses (j>i only)
    bool ki = ((keepw[i >> 4] >> (i & 15)) & 1) != 0;
    __syncthreads();
    if (ki) keepw[t] = (unsigned short)(keepw[t] & (unsigned short)~mat[i * NTILE + t]);
    __syncthreads();
  }
  const unsigned short kw = keepw[t];
  for (int b = 0; b < 16; ++b) {
    int j = t * 16 + b;
    unsigned o = order[j];
    out[o] = scores[o] * (((kw >> b) & 1) ? 1.f : 0.f);
  }
}

// ---------------------------------------------------------------- launcher
extern "C" void kernel_launch(void* const* d_in, const int* in_sizes, int n_in,
                              void* d_out, int out_size, void* d_ws, size_t ws_size,
                              hipStream_t stream) {
  const float* boxes  = (const float*)d_in[0];
  const float* scores = (const float*)d_in[1];
  const int*   labels = (const int*)d_in[2];
  const float* thr    = (const float*)d_in[3];
  float* out = (float*)d_out;
  char* ws = (char*)d_ws;

  unsigned*       order = (unsigned*)(ws + OFF_ORDER);
  float*          cxs   = (float*)(ws + OFF_CX);
  float*          cys   = (float*)(ws + OFF_CY);
  float*          nrm   = (float*)(ws + OFF_NORM);
  float*          rad   = (float*)(ws + OFF_RAD);
  float*          area  = (float*)(ws + OFF_AREA);
  int*            lab   = (int*)(ws + OFF_LAB);
  float*          corn  = (float*)(ws + OFF_CORN);
  unsigned short* mat   = (unsigned short*)(ws + OFF_MAT);

  (void)in_sizes; (void)n_in; (void)out_size; (void)ws_size;

  hipMemsetAsync(mat, 0, (size_t)NB * NTILE * sizeof(unsigned short), stream);
  nms_sort_prep<<<1, 1024, 0, stream>>>(boxes, scores, labels, order, cxs, cys,
                                        nrm, rad, area, lab, corn);
  nms_pairs<<<dim3(NTILE, NTILE), 32, 0, stream>>>(cxs, cys, nrm, rad, area, lab,
                                                   corn, thr, mat);
  nms_scan<<<1, NTILE, 0, stream>>>(mat, order, scores, out);
}